// get_offsetmap_12317966205150
// MI455X (gfx1250) — compile-verified
//
#include <hip/hip_runtime.h>
#include <hip/hip_bf16.h>

typedef __attribute__((ext_vector_type(2))) float v2f;
typedef __attribute__((ext_vector_type(4))) float f4;
typedef __attribute__((ext_vector_type(8))) float v8f;

#define TROWS 21
#define NPTS  1024
#define KSEL  64

__global__ __launch_bounds__(256) void knn_mask_offsetmap_kernel(
    const float* __restrict__ pc,   // (B, 1024, 3)
    const float* __restrict__ tg,   // (B, 21, 3)
    float* __restrict__ out)        // (B, 21, 1024, 3)
{
    // ---- LDS (~148 KB of the 320 KB WGP pool) ----
    // sD is padded to 32 rows so WMMA D-tile stores need no row guard.
    // Its first 12 KB double as the async-load staging area for the raw pc block.
    __shared__ __attribute__((aligned(16))) float sD[32 * NPTS];  // 131072 B
    __shared__ float sPx[NPTS];
    __shared__ float sPy[NPTS];
    __shared__ float sPz[NPTS];
    __shared__ float sPn[NPTS];          // ||p||^2
    __shared__ float sT[32 * 4];         // target rows padded to 32 x 4 (zeros)
    __shared__ float sTn[32];            // ||t||^2 padded

    const int b    = blockIdx.x;
    const int tid  = threadIdx.x;
    const int lane = tid & 31;
    const int wave = tid >> 5;
    const int half = lane >> 4;      // 0 = lanes 0-15, 1 = lanes 16-31
    const int l15  = lane & 15;

    const float* pcb = pc + (size_t)b * NPTS * 3;
    const float* tgb = tg + (size_t)b * TROWS * 3;

    __builtin_prefetch(tgb, 0, 0);   // global_prefetch_b8

    // ---- Phase 1a: async-copy raw pc block (12 KB) global -> LDS staging ----
    // GVS mode: saddr = pc base (SGPR pair), vaddr = 32-bit byte offset,
    // vdst = LDS byte address. Tracked by ASYNCcnt.
    {
        const unsigned sBase = (unsigned)(uintptr_t)(&sD[0]);
#pragma unroll
        for (int it = 0; it < 3; ++it) {
            unsigned goff = (unsigned)(tid + it * 256) * 16u;  // 768 x 16B = 12288B
            unsigned lds  = sBase + goff;
            asm volatile("global_load_async_to_lds_b128 %0, %1, %2 offset:0"
                         :
                         : "v"(lds), "v"(goff), "s"(pcb)
                         : "memory");
        }
        asm volatile("s_wait_asynccnt 0" ::: "memory");
    }
    // target rows (tiny) via normal loads
    if (tid < 32) {
        float x = 0.f, y = 0.f, z = 0.f;
        if (tid < TROWS) {
            x = tgb[tid * 3 + 0];
            y = tgb[tid * 3 + 1];
            z = tgb[tid * 3 + 2];
        }
        sT[tid * 4 + 0] = x;
        sT[tid * 4 + 1] = y;
        sT[tid * 4 + 2] = z;
        sT[tid * 4 + 3] = 0.f;
        sTn[tid] = x * x + y * y + z * z;
    }
    __syncthreads();

    // ---- Phase 1b: unpack staged AoS pc into SoA planes + norms ----
    for (int i = tid; i < NPTS; i += 256) {
        float x = sD[i * 3 + 0];
        float y = sD[i * 3 + 1];
        float z = sD[i * 3 + 2];
        sPx[i] = x; sPy[i] = y; sPz[i] = z;
        sPn[i] = x * x + y * y + z * z;
    }
    __syncthreads();

    // ---- Phase 2: D = (-2*T) * P^T + (||t||^2 + ||p||^2) via WMMA f32 16x16x4 ----
    // mt outer so the A fragment + row-norm terms are hoisted; all fragment
    // construction is branchless (unconditional LDS loads + cndmask selects).
#pragma unroll
    for (int mt = 0; mt < 2; ++mt) {
        const int mrow = l15 + 16 * mt;       // A row this lane holds (padded -> 0)
        const float tk0 = sT[mrow * 4 + 0];
        const float tk1 = sT[mrow * 4 + 1];
        const float tk2 = sT[mrow * 4 + 2];

        // A fragment (16x4): VGPR0 = K0 (lanes 0-15) / K2 (lanes 16-31); VGPR1 = K1 / K3(=0)
        v2f a;
        a.x = -2.0f * (half ? tk2 : tk0);
        a.y = half ? 0.0f : (-2.0f * tk1);

        float tn[8];
#pragma unroll
        for (int r = 0; r < 8; ++r)
            tn[r] = sTn[mt * 16 + r + 8 * half];  // ||t||^2 for M = mt*16 + r + 8*half

        for (int nt = wave; nt < 64; nt += 8) {
            const int n = nt * 16 + l15;      // column this lane handles in B/C/D

            const float px = sPx[n];
            const float py = sPy[n];
            const float pz = sPz[n];
            const float pn = sPn[n];

            // B fragment (4x16): VGPR0 = row K0 / K2; VGPR1 = row K1 / K3(=0)
            v2f bf;
            bf.x = half ? pz : px;
            bf.y = half ? 0.0f : py;

            // C fragment: VGPR r -> M = mt*16 + r + 8*half, N = n
            v8f c;
#pragma unroll
            for (int r = 0; r < 8; ++r)
                c[r] = tn[r] + pn;

            v8f dacc = __builtin_amdgcn_wmma_f32_16x16x4_f32(
                false, a, false, bf, (short)0, c, false, false);

            // unconditional stores into the 32-row padded matrix
#pragma unroll
            for (int r = 0; r < 8; ++r) {
                const int m = mt * 16 + r + 8 * half;
                sD[m * NPTS + n] = dacc[r];
            }
        }
    }
    __syncthreads();

    // ---- Phase 3: per-row exact top-64 (radix select) + masked b128 output stream ----
    for (int t = wave; t < TROWS; t += 8) {
        const int base = t * NPTS + lane * 32;

        // order-preserving float -> uint keys, cached in registers
        unsigned int u[32];
#pragma unroll
        for (int j = 0; j < 32; ++j) {
            unsigned int bits = __float_as_uint(sD[base + j]);
            unsigned int flip = (unsigned int)((int)bits >> 31) | 0x80000000u;
            u[j] = bits ^ flip;
        }

        // MSB-first radix select: exact K-th smallest key across the wave
        unsigned int prefix = 0;
        int need = KSEL;
        for (int bit = 31; bit >= 0; --bit) {
            const unsigned int p = prefix >> bit;
            int cnt = 0;
#pragma unroll
            for (int j = 0; j < 32; ++j)
                cnt += ((u[j] >> bit) == p) ? 1 : 0;
            for (int off = 16; off; off >>= 1)
                cnt += __shfl_xor(cnt, off, 32);
            if (cnt < need) {
                need -= cnt;
                prefix |= (1u << bit);
            }
        }
        const unsigned int thresh = prefix;  // K-th smallest key
        const int eq_take = need;            // # of equal keys to take (index order)

        // stable rank among equal keys: wave exclusive scan of per-lane equal counts
        int eq_in = 0;
#pragma unroll
        for (int j = 0; j < 32; ++j)
            eq_in += (u[j] == thresh) ? 1 : 0;
        int scan = eq_in;
        for (int off = 1; off < 32; off <<= 1) {
            int v = __shfl_up(scan, off, 32);
            if (lane >= off) scan += v;
        }
        int eqrank = scan - eq_in;           // exclusive prefix

        // each lane streams 96 contiguous floats = 24 x global_store_b128
        float* orow = out + ((size_t)(b * TROWS + t) * NPTS + lane * 32) * 3;
        f4* orow4 = reinterpret_cast<f4*>(orow);
#pragma unroll
        for (int g = 0; g < 8; ++g) {
            float v[12];
#pragma unroll
            for (int q = 0; q < 4; ++q) {
                const int j = g * 4 + q;
                bool sel;
                if (u[j] < thresh) {
                    sel = true;
                } else if (u[j] == thresh) {
                    sel = (eqrank < eq_take);
                    eqrank++;
                } else {
                    sel = false;
                }
                const int n = lane * 32 + j;
                v[q * 3 + 0] = sel ? sPx[n] : 0.0f;
                v[q * 3 + 1] = sel ? sPy[n] : 0.0f;
                v[q * 3 + 2] = sel ? sPz[n] : 0.0f;
            }
            orow4[g * 3 + 0] = f4{v[0], v[1], v[2],  v[3]};
            orow4[g * 3 + 1] = f4{v[4], v[5], v[6],  v[7]};
            orow4[g * 3 + 2] = f4{v[8], v[9], v[10], v[11]};
        }
    }
}

extern "C" void kernel_launch(void* const* d_in, const int* in_sizes, int n_in,
                              void* d_out, int out_size, void* d_ws, size_t ws_size,
                              hipStream_t stream) {
    const float* pc = (const float*)d_in[0];   // (B,1024,3) f32
    const float* tg = (const float*)d_in[1];   // (B,21,3)   f32
    float* out = (float*)d_out;                // (B,21,1024,3) f32

    const int B = in_sizes[0] / (NPTS * 3);    // 512
    knn_mask_offsetmap_kernel<<<dim3(B), dim3(256), 0, stream>>>(pc, tg, out);
}